// AccelerationTorch_68375879352859
// MI455X (gfx1250) — compile-verified
//
#include <hip/hip_runtime.h>
#include <stdint.h>

// ---------------------------------------------------------------------------
// Tiling constants
// ---------------------------------------------------------------------------
#define LANES            32                      // gfx1250 is wave32
#define EPT              4                       // edges per lane per tile
#define TILE_EDGES       (LANES * EPT)           // 128 edges per wave-tile
#define WAVES_PER_BLOCK  8
#define BLOCK_THREADS    (WAVES_PER_BLOCK * LANES)   // 256

#define ROWS_BYTES       (TILE_EDGES * 8)        // 1024 B of int64 row idx
#define COLS_BYTES       (TILE_EDGES * 8)        // 1024 B of int64 col idx
#define K_BYTES          (TILE_EDGES * 4)        //  512 B of f32 K
#define BUF_BYTES        (ROWS_BYTES + COLS_BYTES + K_BYTES)   // 2560 B

// ---------------------------------------------------------------------------
// CDNA5 async global->LDS copy (16 B per lane per issue), tracked by ASYNCcnt.
// Non-temporal hint: the edge stream is touched exactly once, so keep it from
// displacing the L2-resident phase[] / accumulator[] arrays (hot: 32M refs).
// ---------------------------------------------------------------------------
__device__ __forceinline__ void async_copy_b128(unsigned lds_addr, const void* gaddr) {
    asm volatile("global_load_async_to_lds_b128 %0, %1, off th:TH_LOAD_NT"
                 :
                 : "v"(lds_addr), "v"(gaddr)
                 : "memory");
}

// Stage one 128-edge tile (rows, cols, K) into this wave's LDS buffer.
// 5 async-load instructions per wave, each a coalesced 512 B chunk.
__device__ __forceinline__ void stage_tile(unsigned lds_base, int lane,
                                           const char* g_rows,
                                           const char* g_cols,
                                           const char* g_k) {
    const unsigned o = (unsigned)lane * 16u;
    async_copy_b128(lds_base + o,                               g_rows + o);
    async_copy_b128(lds_base + o + 512u,                        g_rows + o + 512);
    async_copy_b128(lds_base + ROWS_BYTES + o,                  g_cols + o);
    async_copy_b128(lds_base + ROWS_BYTES + o + 512u,           g_cols + o + 512);
    async_copy_b128(lds_base + ROWS_BYTES + COLS_BYTES + o,     g_k    + o);
}

// ---------------------------------------------------------------------------
// Kernel 1: out[i] = power[i] - gamma[i]*dphase[i]
// ---------------------------------------------------------------------------
__global__ void node_init_kernel(const float* __restrict__ power,
                                 const float* __restrict__ gamma,
                                 const float* __restrict__ dphase,
                                 float* __restrict__ out, int n) {
    int i = blockIdx.x * blockDim.x + threadIdx.x;
    if (i < n) out[i] = fmaf(-gamma[i], dphase[i], power[i]);
}

// ---------------------------------------------------------------------------
// Kernel 2: double-buffered async-staged edge scatter.
// One wave per 128-edge tile, grid-strided over tiles. Wave-uniform control
// flow keeps EXEC all-ones for the async ops.
// ---------------------------------------------------------------------------
__global__ void edge_scatter_kernel(const float* __restrict__ phase,
                                    const float* __restrict__ Kv,
                                    const long long* __restrict__ row,
                                    const long long* __restrict__ col,
                                    float* __restrict__ out,
                                    long long num_tiles) {
    __shared__ __align__(16) unsigned char smem[WAVES_PER_BLOCK * 2 * BUF_BYTES];

    const int lane = threadIdx.x & (LANES - 1);
    const int wave = threadIdx.x >> 5;
    const long long gwave  = (long long)blockIdx.x * WAVES_PER_BLOCK + wave;
    const long long nwaves = (long long)gridDim.x * WAVES_PER_BLOCK;

    unsigned char* wave_smem = smem + (size_t)wave * (2 * BUF_BYTES);
    // Low 32 bits of a generic LDS pointer are the LDS byte address.
    const unsigned lds_base0 = (unsigned)(uintptr_t)wave_smem;

    long long tile = gwave;
    if (tile >= num_tiles) return;

    // Prime the pipeline: stage first tile into buffer 0.
    stage_tile(lds_base0, lane,
               (const char*)(row + tile * TILE_EDGES),
               (const char*)(col + tile * TILE_EDGES),
               (const char*)(Kv  + tile * TILE_EDGES));

    int cur = 0;
    for (;;) {
        const long long next_tile = tile + nwaves;
        const bool has_next = (next_tile < num_tiles);

        if (has_next) {
            // Stage next tile into the other buffer, then wait until only its
            // 5 async loads remain outstanding. Async loads complete in order,
            // so ASYNCcnt <= 5 implies the current tile's data is in LDS.
            stage_tile(lds_base0 + (unsigned)(cur ^ 1) * BUF_BYTES, lane,
                       (const char*)(row + next_tile * TILE_EDGES),
                       (const char*)(col + next_tile * TILE_EDGES),
                       (const char*)(Kv  + next_tile * TILE_EDGES));
            asm volatile("s_wait_asynccnt 0x5" ::: "memory");
        } else {
            asm volatile("s_wait_asynccnt 0x0" ::: "memory");
        }

        // ---- consume current buffer (lane-strided: bank-conflict-free) ----
        const unsigned char* buf = wave_smem + (size_t)cur * BUF_BYTES;
        const long long* rr = (const long long*)(buf);
        const long long* cc = (const long long*)(buf + ROWS_BYTES);
        const float*     kk = (const float*)(buf + ROWS_BYTES + COLS_BYTES);

#pragma unroll
        for (int j = 0; j < EPT; ++j) {
            const int e  = lane + LANES * j;
            const int ri = (int)rr[e];
            const int ci = (int)cc[e];
            const float k = kk[e];
            const float s = k * sinf(phase[ci] - phase[ri]);  // L2-resident gathers
            atomicAdd(out + ri,  s);                          // global_atomic_add_f32
            atomicAdd(out + ci, -s);
        }

        if (!has_next) break;
        tile = next_tile;
        cur ^= 1;
    }
}

// ---------------------------------------------------------------------------
// Tail kernel for E % TILE_EDGES edges (0 for E = 16M, but stay general)
// ---------------------------------------------------------------------------
__global__ void edge_tail_kernel(const float* __restrict__ phase,
                                 const float* __restrict__ Kv,
                                 const long long* __restrict__ row,
                                 const long long* __restrict__ col,
                                 float* __restrict__ out,
                                 long long start, long long E) {
    long long i = start + (long long)blockIdx.x * blockDim.x + threadIdx.x;
    if (i < E) {
        const int ri = (int)row[i];
        const int ci = (int)col[i];
        const float s = Kv[i] * sinf(phase[ci] - phase[ri]);
        atomicAdd(out + ri,  s);
        atomicAdd(out + ci, -s);
    }
}

// ---------------------------------------------------------------------------
// Kernel 3: out[i] /= mass[i]
// ---------------------------------------------------------------------------
__global__ void node_finish_kernel(const float* __restrict__ mass,
                                   float* __restrict__ out, int n) {
    int i = blockIdx.x * blockDim.x + threadIdx.x;
    if (i < n) out[i] = out[i] / mass[i];
}

// ---------------------------------------------------------------------------
// Launcher
// ---------------------------------------------------------------------------
extern "C" void kernel_launch(void* const* d_in, const int* in_sizes, int n_in,
                              void* d_out, int out_size, void* d_ws, size_t ws_size,
                              hipStream_t stream) {
    (void)n_in; (void)out_size; (void)d_ws; (void)ws_size;

    const float*     phase  = (const float*)d_in[0];
    const float*     dphase = (const float*)d_in[1];
    const float*     power  = (const float*)d_in[2];
    const float*     mass   = (const float*)d_in[3];
    const float*     gamma  = (const float*)d_in[4];
    const float*     Kv     = (const float*)d_in[5];
    const long long* ei     = (const long long*)d_in[6];   // int64 indices, shape (2,E)

    const int       n = in_sizes[0];
    const long long E = (long long)in_sizes[5];
    const long long* row = ei;
    const long long* col = ei + E;
    float* out = (float*)d_out;

    // 1) base acceleration
    node_init_kernel<<<(n + 255) / 256, 256, 0, stream>>>(power, gamma, dphase, out, n);

    // 2) edge scatter (async double-buffered main path + scalar tail)
    const long long num_tiles = E / TILE_EDGES;
    if (num_tiles > 0) {
        long long blocks_needed = (num_tiles + WAVES_PER_BLOCK - 1) / WAVES_PER_BLOCK;
        int blocks = (blocks_needed < 2048) ? (int)blocks_needed : 2048;
        edge_scatter_kernel<<<blocks, BLOCK_THREADS, 0, stream>>>(
            phase, Kv, row, col, out, num_tiles);
    }
    const long long tail_start = num_tiles * TILE_EDGES;
    const long long rem = E - tail_start;
    if (rem > 0) {
        edge_tail_kernel<<<(int)((rem + 255) / 256), 256, 0, stream>>>(
            phase, Kv, row, col, out, tail_start, E);
    }

    // 3) divide by mass
    node_finish_kernel<<<(n + 255) / 256, 256, 0, stream>>>(mass, out, n);
}